// GCNModel_60447369723924
// MI455X (gfx1250) — compile-verified
//
#include <hip/hip_runtime.h>

// ---------------------------------------------------------------------------
// 2-layer GCN for MI455X (gfx1250, wave32).
//   layer: agg = D^-1/2 (A+I) D^-1/2 (x @ W);  out = act(agg + b)
// GEMMs use V_WMMA_F32_16X16X4_F32 (full fp32 precision, matches reference).
// Edge aggregation: wave-per-edge gather + global_atomic_add_f32 scatter
// (L2-resident working set on the 192MB L2; HBM sees ~one compulsory pass).
// ---------------------------------------------------------------------------

typedef __attribute__((ext_vector_type(2))) float v2f;
typedef __attribute__((ext_vector_type(8))) float v8f;

__global__ void fill_kernel(float* __restrict__ p, float v, int n) {
  int t = blockIdx.x * blockDim.x + threadIdx.x;
  if (t < n) p[t] = v;
}

// deg[dst] += 1 for every edge (deg pre-initialized to 1.0 for self loops)
__global__ void deg_count_kernel(const int* __restrict__ ei, float* __restrict__ deg, int E) {
  int t = blockIdx.x * blockDim.x + threadIdx.x;
  if (t < E) atomicAdd(&deg[ei[E + t]], 1.0f);
}

__global__ void rsqrt_kernel(float* __restrict__ p, int n) {
  int t = blockIdx.x * blockDim.x + threadIdx.x;
  if (t < n) p[t] = rsqrtf(p[t]);
}

// C[M,N] = A[M,K] @ B[K,N], N = NT*16, M % 16 == 0, K % 4 == 0.
// One wave computes a 16-row x N panel (A fragment reused across all N tiles).
template <int NT>
__global__ __launch_bounds__(256) void wmma_gemm_f32_kernel(
    const float* __restrict__ A, const float* __restrict__ B,
    float* __restrict__ C, int M, int K) {
  constexpr int N = NT * 16;
  const int w    = blockIdx.x * (blockDim.x >> 5) + (threadIdx.x >> 5);
  const int lane = threadIdx.x & 31;
  if (w * 16 >= M) return;  // uniform per-wave exit (EXEC stays all-1 for WMMA)

  // A 16x4 f32 fragment: lanes 0-15 -> K={0,1}, lanes 16-31 -> K={2,3}
  const int arow = w * 16 + (lane & 15);
  const int kb   = (lane >> 4) << 1;   // 0 or 2
  const int bcol = lane & 15;          // B/C column within tile

  v8f acc[NT] = {};

  const float* ap = A + (size_t)arow * K + kb;
  for (int k = 0; k < K; k += 4) {
    v2f a;
    a.x = ap[k];
    a.y = ap[k + 1];
#pragma unroll
    for (int t = 0; t < NT; ++t) {
      const float* bp = B + (size_t)(k + kb) * N + t * 16 + bcol;
      v2f b;
      b.x = bp[0];  // K = k+kb   row of B
      b.y = bp[N];  // K = k+kb+1 row of B
      // 8 args: (neg_a, A, neg_b, B, c_mod, C, reuse_a, reuse_b)
      acc[t] = __builtin_amdgcn_wmma_f32_16x16x4_f32(
          false, a, false, b, (short)0, acc[t], false, false);
    }
  }

  // C/D layout: VGPR r -> M = base + (lane>=16 ? 8 : 0) + r, N = lane&15
  const int crow = w * 16 + ((lane >> 4) << 3);
#pragma unroll
  for (int t = 0; t < NT; ++t) {
    float* cp = C + (size_t)crow * N + t * 16 + bcol;
#pragma unroll
    for (int r = 0; r < 8; ++r) cp[(size_t)r * N] = acc[t][r];
  }
}

// out[i,:] = hg[i,:] * dinv[i]^2   (self-loop term; also zero-inits agg buffer)
__global__ void selfloop_kernel(const float* __restrict__ hg, const float* __restrict__ dinv,
                                float* __restrict__ out, int M, int dshift) {
  int t = blockIdx.x * blockDim.x + threadIdx.x;
  int total = M << dshift;
  if (t >= total) return;
  int i = t >> dshift;
  float di = dinv[i];
  out[t] = hg[t] * (di * di);
}

// wave-per-edge: out[dst,:] += h[src,:] * dinv[src]*dinv[dst]
__global__ void edge_scatter_kernel(const float* __restrict__ h, const int* __restrict__ ei,
                                    const float* __restrict__ dinv, float* __restrict__ out,
                                    int E, int dshift) {
  int w = blockIdx.x * (blockDim.x >> 5) + (threadIdx.x >> 5);
  if (w >= E) return;
  int lane = threadIdx.x & 31;
  int s = ei[w];       // edge_index[0] = message source
  int q = ei[E + w];   // edge_index[1] = aggregation target
  float norm = dinv[s] * dinv[q];
  int D = 1 << dshift;
  const float* hp = h + ((size_t)s << dshift);
  float* op = out + ((size_t)q << dshift);
  for (int d = lane; d < D; d += 32)
    atomicAdd(op + d, hp[d] * norm);
}

__global__ void bias_kernel(float* __restrict__ out, const float* __restrict__ b,
                            int total, int mask, int relu) {
  int t = blockIdx.x * blockDim.x + threadIdx.x;
  if (t >= total) return;
  float v = out[t] + b[t & mask];
  out[t] = relu ? fmaxf(v, 0.0f) : v;
}

extern "C" void kernel_launch(void* const* d_in, const int* in_sizes, int n_in,
                              void* d_out, int out_size, void* d_ws, size_t ws_size,
                              hipStream_t stream) {
  const float* x  = (const float*)d_in[0];
  const int*   ei = (const int*)d_in[1];
  const float* W1 = (const float*)d_in[2];
  const float* b1 = (const float*)d_in[3];
  const float* W2 = (const float*)d_in[4];
  const float* b2 = (const float*)d_in[5];
  float* out = (float*)d_out;

  const int M  = 50000;  // N_NODES (M % 16 == 0)
  const int K1 = 256, D1 = 128, D2 = 64;
  const int E  = in_sizes[1] / 2;

  // workspace layout
  char* ws = (char*)d_ws;
  size_t off = 0;
  float* dinv = (float*)(ws + off); off += ((size_t)M * 4 + 255) & ~(size_t)255;
  float* hg1  = (float*)(ws + off); off += ((size_t)M * D1 * 4 + 255) & ~(size_t)255;
  float* ag1  = (float*)(ws + off); off += ((size_t)M * D1 * 4 + 255) & ~(size_t)255;
  float* hg2  = (float*)(ws + off);

  const int TB = 256;
  const int wavesPerBlk = TB / 32;
  const int gemmWaves = M / 16;  // 3125
  const int gemmBlks  = (gemmWaves + wavesPerBlk - 1) / wavesPerBlk;
  const int edgeBlks  = (E + wavesPerBlk - 1) / wavesPerBlk;

  // --- degrees -> dinv ---
  fill_kernel<<<(M + TB - 1) / TB, TB, 0, stream>>>(dinv, 1.0f, M);
  deg_count_kernel<<<(E + TB - 1) / TB, TB, 0, stream>>>(ei, dinv, E);
  rsqrt_kernel<<<(M + TB - 1) / TB, TB, 0, stream>>>(dinv, M);

  // --- layer 1 ---
  wmma_gemm_f32_kernel<D1 / 16><<<gemmBlks, TB, 0, stream>>>(x, W1, hg1, M, K1);
  selfloop_kernel<<<((M * D1) + TB - 1) / TB, TB, 0, stream>>>(hg1, dinv, ag1, M, 7);
  edge_scatter_kernel<<<edgeBlks, TB, 0, stream>>>(hg1, ei, dinv, ag1, E, 7);
  bias_kernel<<<((M * D1) + TB - 1) / TB, TB, 0, stream>>>(ag1, b1, M * D1, D1 - 1, 1);

  // --- layer 2 (aggregates directly into d_out) ---
  wmma_gemm_f32_kernel<D2 / 16><<<gemmBlks, TB, 0, stream>>>(ag1, W2, hg2, M, D1);
  selfloop_kernel<<<((M * D2) + TB - 1) / TB, TB, 0, stream>>>(hg2, dinv, out, M, 6);
  edge_scatter_kernel<<<edgeBlks, TB, 0, stream>>>(hg2, ei, dinv, out, E, 6);
  bias_kernel<<<((M * D2) + TB - 1) / TB, TB, 0, stream>>>(out, b2, M * D2, D2 - 1, 0);
}